// GraphConv_14001593385076
// MI455X (gfx1250) — compile-verified
//
#include <hip/hip_runtime.h>
#include <hip/hip_bf16.h>

#define N_NODES 100000
#define N_EDGES 1600000
#define IN_F    256
#define OUT_F   128
#define N_STRIPES (N_NODES / 16)   // 6250, N_NODES divisible by 16

typedef __attribute__((ext_vector_type(2))) float v2f;
typedef __attribute__((ext_vector_type(8))) float v8f;

// ---------------------------------------------------------------------------
// 1) degree accumulation (deterministic integer atomics)
// ---------------------------------------------------------------------------
__global__ void gc_deg_kernel(const int* __restrict__ dst, unsigned int* __restrict__ deg) {
    int e = blockIdx.x * blockDim.x + threadIdx.x;
    if (e < N_EDGES) atomicAdd(&deg[dst[e]], 1u);
}

// 2) norm[n] = rsqrt(max(deg,1))  (in-place uint -> float)
__global__ void gc_norm_kernel(float* __restrict__ norm_buf) {
    int n = blockIdx.x * blockDim.x + threadIdx.x;
    if (n < N_NODES) {
        unsigned int d = reinterpret_cast<unsigned int*>(norm_buf)[n];
        float df = (d < 1u) ? 1.0f : (float)d;
        norm_buf[n] = rsqrtf(df);
    }
}

// ---------------------------------------------------------------------------
// 3) h = (feat * norm[:,None]) @ weight  via V_WMMA_F32_16X16X4_F32
//    One wave -> one 16-row x 128-col stripe (8 accumulator tiles).
//    A layout (16x4 f32): lanes 0-15 hold K={0,1}, lanes 16-31 hold K={2,3},
//    row M = lane&15. The two A floats per lane are memory-contiguous.
//    B layout (4x16 f32): lanes 0-15 hold K={0,1}, lanes 16-31 K={2,3},
//    col N = lane&15, one K per VGPR.
// ---------------------------------------------------------------------------
__global__ __launch_bounds__(256) void gc_gemm_kernel(
        const float* __restrict__ feat,
        const float* __restrict__ weight,
        const float* __restrict__ norm,
        float* __restrict__ h) {
    const int lane   = threadIdx.x & 31;
    const int wave   = threadIdx.x >> 5;
    const int stripe = blockIdx.x * 8 + wave;
    if (stripe >= N_STRIPES) return;

    const int row0  = stripe * 16;
    const int m     = lane & 15;          // row within tile / col within B tile
    const int khalf = (lane >> 4) << 1;   // 0 for lanes 0-15, 2 for lanes 16-31

    const float nm = norm[row0 + m];      // fold norm scale into A fragment
    const float* __restrict__ arow = feat + (size_t)(row0 + m) * IN_F + khalf;

    v8f acc[8];
    const v8f vzero = {0.f,0.f,0.f,0.f,0.f,0.f,0.f,0.f};
#pragma unroll
    for (int n = 0; n < 8; ++n) acc[n] = vzero;

    for (int kk = 0; kk < IN_F; kk += 4) {
        // A fragment: contiguous float2 per lane, pre-scaled by norm
        const float2 af = *reinterpret_cast<const float2*>(arow + kk);
        v2f a;
        a.x = af.x * nm;
        a.y = af.y * nm;

        const float* __restrict__ brow0 = weight + (size_t)(kk + khalf) * OUT_F;
        const float* __restrict__ brow1 = brow0 + OUT_F;
#pragma unroll
        for (int n = 0; n < 8; ++n) {
            v2f b;
            b.x = brow0[n * 16 + m];
            b.y = brow1[n * 16 + m];
            // (neg_a, A, neg_b, B, c_mod, C, reuse_a, reuse_b)
            acc[n] = __builtin_amdgcn_wmma_f32_16x16x4_f32(
                false, a, false, b, (short)0, acc[n], false, false);
        }
    }

    // D layout (16x16 f32): VGPR v, lanes 0-15 -> M=v, N=lane; lanes 16-31 -> M=v+8
    const int ncol  = lane & 15;
    const int mbase = (lane >> 4) * 8;
#pragma unroll
    for (int n = 0; n < 8; ++n) {
#pragma unroll
        for (int v = 0; v < 8; ++v) {
            h[(size_t)(row0 + mbase + v) * OUT_F + n * 16 + ncol] = acc[n][v];
        }
    }
}

// ---------------------------------------------------------------------------
// 4) edge scatter: out[dst] += h[src]; one wave per edge, float4 per lane.
//    h and out both fit in the 192 MB L2, so random traffic stays on-chip.
// ---------------------------------------------------------------------------
__global__ __launch_bounds__(256) void gc_scatter_kernel(
        const int* __restrict__ src, const int* __restrict__ dst,
        const float* __restrict__ h, float* __restrict__ out) {
    const int e = blockIdx.x * 8 + (threadIdx.x >> 5);
    if (e >= N_EDGES) return;
    const int lane = threadIdx.x & 31;
    const int s = src[e];
    const int d = dst[e];
    const float4 v = *reinterpret_cast<const float4*>(h + (size_t)s * OUT_F + lane * 4);
    float* o = out + (size_t)d * OUT_F + lane * 4;
    atomicAdd(o + 0, v.x);
    atomicAdd(o + 1, v.y);
    atomicAdd(o + 2, v.z);
    atomicAdd(o + 3, v.w);
}

// 5) finalize: out = out * norm[node] + bias[feat]   (in-place, streaming)
__global__ void gc_final_kernel(float* __restrict__ out,
                                const float* __restrict__ norm,
                                const float* __restrict__ bias) {
    const long long i = (long long)blockIdx.x * blockDim.x + threadIdx.x;
    if (i < (long long)N_NODES * OUT_F) {
        const int n = (int)(i >> 7);
        const int f = (int)(i & 127);
        out[i] = out[i] * norm[n] + bias[f];
    }
}

// ---------------------------------------------------------------------------
extern "C" void kernel_launch(void* const* d_in, const int* in_sizes, int n_in,
                              void* d_out, int out_size, void* d_ws, size_t ws_size,
                              hipStream_t stream) {
    const float* feat   = (const float*)d_in[0];
    const float* weight = (const float*)d_in[1];
    const float* bias   = (const float*)d_in[2];
    const int*   src    = (const int*)d_in[3];
    const int*   dst    = (const int*)d_in[4];
    float*       out    = (float*)d_out;

    // workspace layout: [norm: 100096 floats (padded)] [h: N_NODES*OUT_F floats]
    float* norm = (float*)d_ws;
    float* h    = norm + 100096;

    // zero degree buffer and output accumulator (stream ops: graph-capturable)
    hipMemsetAsync(norm, 0, (size_t)N_NODES * sizeof(float), stream);
    hipMemsetAsync(out, 0, (size_t)N_NODES * OUT_F * sizeof(float), stream);

    gc_deg_kernel<<<(N_EDGES + 255) / 256, 256, 0, stream>>>(dst, (unsigned int*)norm);
    gc_norm_kernel<<<(N_NODES + 255) / 256, 256, 0, stream>>>(norm);
    gc_gemm_kernel<<<(N_STRIPES + 7) / 8, 256, 0, stream>>>(feat, weight, norm, h);
    gc_scatter_kernel<<<N_EDGES / 8, 256, 0, stream>>>(src, dst, h, out);
    gc_final_kernel<<<((long long)N_NODES * OUT_F + 255) / 256, 256, 0, stream>>>(out, norm, bias);
}